// AttentionLayer_64785286693309
// MI455X (gfx1250) — compile-verified
//
#include <hip/hip_runtime.h>
#include <hip/hip_bf16.h>
#include <math.h>

typedef __attribute__((ext_vector_type(2))) float v2f;
typedef __attribute__((ext_vector_type(8))) float v8f;

#define BB 4
#define SS 2048
#define DD 1024
#define MTOT (BB * SS)          // 8192 rows
#define SCALE 0.03125f          // 1/sqrt(1024)

// ---------------------------------------------------------------------------
// Kernel 1: fused QKV projection.  One wave computes one 16x16 output tile
// for Q, K and V simultaneously (A fragment loaded once, reused 3x).
// grid = (DD/16, MTOT/16), block = 32 (one wave32).
// ---------------------------------------------------------------------------
__global__ void qkv_proj_kernel(const float* __restrict__ seq,
                                const float* __restrict__ Wq,
                                const float* __restrict__ Wk,
                                const float* __restrict__ Wv,
                                float* __restrict__ Q,
                                float* __restrict__ K,
                                float* __restrict__ V) {
    const int lane  = threadIdx.x;        // 0..31
    const int m16   = lane & 15;          // row within tile (A) / col within tile (B,D)
    const int kh    = lane >> 4;          // K-half select (0/1)
    const int ntile = blockIdx.x;         // 0..63
    const int mtile = blockIdx.y;         // 0..511

    const int   col  = ntile * 16 + m16;              // output / B column
    const float* arow = seq + (size_t)(mtile * 16 + m16) * DD;

    v8f accq = {};
    v8f acck = {};
    v8f accv = {};

    #pragma unroll 4
    for (int k0 = 0; k0 < DD; k0 += 4) {
        const int ka = k0 + 2 * kh;                   // A/B K index for this half
        // A fragment: a[v] = seq[row][ka+v]  (even offset -> aligned b64 load)
        v2f a = *(const v2f*)(arow + ka);
        // B fragments: b[v] = W[ka+v][col]
        v2f bq, bk, bv;
        bq[0] = Wq[(size_t)ka * DD + col];
        bq[1] = Wq[(size_t)(ka + 1) * DD + col];
        bk[0] = Wk[(size_t)ka * DD + col];
        bk[1] = Wk[(size_t)(ka + 1) * DD + col];
        bv[0] = Wv[(size_t)ka * DD + col];
        bv[1] = Wv[(size_t)(ka + 1) * DD + col];

        accq = __builtin_amdgcn_wmma_f32_16x16x4_f32(false, a, false, bq, (short)0, accq, false, false);
        acck = __builtin_amdgcn_wmma_f32_16x16x4_f32(false, a, false, bk, (short)0, acck, false, false);
        accv = __builtin_amdgcn_wmma_f32_16x16x4_f32(false, a, false, bv, (short)0, accv, false, false);
    }

    // D layout: VGPR r, lanes 0-15 -> row r, lanes 16-31 -> row r+8; col = lane%16
    #pragma unroll
    for (int r = 0; r < 8; ++r) {
        const size_t orow = (size_t)(mtile * 16 + r + 8 * kh);
        Q[orow * DD + col] = accq[r];
        K[orow * DD + col] = acck[r];
        V[orow * DD + col] = accv[r];
    }
}

// ---------------------------------------------------------------------------
// Kernel 2: attention for one (batch, 16-query tile) per wave.
//  Phase 1: scores[16][2048] -> LDS (scaled)
//  Phase 2: softmax in LDS (two half-rows per lane, shfl_xor(16) to combine)
//  Phase 3: P @ V streamed per 16-col chunk, acc in VGPRs, fused epilogue
// grid = (SS/16, BB), block = 32, dynamic LDS = (16*2048 + 16) floats.
// ---------------------------------------------------------------------------
__global__ void attn_kernel(const float* __restrict__ Q,
                            const float* __restrict__ K,
                            const float* __restrict__ V,
                            const float* __restrict__ seq,
                            const float* __restrict__ mask,
                            float* __restrict__ out) {
    extern __shared__ float lds[];
    float* sc = lds;               // [16][SS] score strip
    float* rs = lds + 16 * SS;     // [16] reciprocal row sums

    const int lane = threadIdx.x;
    const int m16  = lane & 15;
    const int kh   = lane >> 4;
    const int qt   = blockIdx.x;           // query tile 0..127
    const int b    = blockIdx.y;           // batch 0..3
    const size_t base = (size_t)b * SS * DD;

    const float* qrow = Q + base + (size_t)(qt * 16 + m16) * DD;

    // ---- Phase 1: scores = (q @ k^T) * scale --------------------------------
    for (int kt = 0; kt < SS / 16; ++kt) {
        const float* krow = K + base + (size_t)(kt * 16 + m16) * DD; // B col = key row
        v8f acc = {};
        #pragma unroll 8
        for (int k0 = 0; k0 < DD; k0 += 4) {
            const int ka = k0 + 2 * kh;
            v2f a  = *(const v2f*)(qrow + ka);   // A: q rows
            v2f bb = *(const v2f*)(krow + ka);   // B: k^T -> contiguous k row
            acc = __builtin_amdgcn_wmma_f32_16x16x4_f32(false, a, false, bb, (short)0, acc, false, false);
        }
        #pragma unroll
        for (int r = 0; r < 8; ++r)
            sc[(r + 8 * kh) * SS + kt * 16 + m16] = acc[r] * SCALE;
    }
    __syncthreads();

    // ---- Phase 2: softmax over each row of the strip ------------------------
    // lane handles row m16, half kh -> columns [kh*1024, kh*1024+1024)
    {
        float* srow = sc + m16 * SS + kh * (SS / 2);
        float mx = -INFINITY;
        for (int i = 0; i < SS / 2; ++i) mx = fmaxf(mx, srow[i]);
        mx = fmaxf(mx, __shfl_xor(mx, 16, 32));          // combine halves
        float sum = 0.0f;
        for (int i = 0; i < SS / 2; ++i) {
            float e = __expf(srow[i] - mx);
            srow[i] = e;
            sum += e;
        }
        sum += __shfl_xor(sum, 16, 32);
        if (kh == 0) rs[m16] = 1.0f / sum;
    }
    __syncthreads();

    // ---- Phase 3: out = seq + mask * (P @ V) / rowsum -----------------------
    for (int nt = 0; nt < DD / 16; ++nt) {
        const int col = nt * 16 + m16;
        v8f acc = {};
        #pragma unroll 8
        for (int kk = 0; kk < SS; kk += 4) {
            const int kr = kk + 2 * kh;
            // A fragment from LDS score strip: a[v] = P[m16][kr+v]
            v2f a = *(const v2f*)(sc + m16 * SS + kr);
            // B fragment: b[v] = V[kr+v][col]
            v2f bb;
            bb[0] = V[base + (size_t)kr * DD + col];
            bb[1] = V[base + (size_t)(kr + 1) * DD + col];
            acc = __builtin_amdgcn_wmma_f32_16x16x4_f32(false, a, false, bb, (short)0, acc, false, false);
        }
        const float mk = mask[col];
        #pragma unroll
        for (int r = 0; r < 8; ++r) {
            const int   row = r + 8 * kh;
            const size_t idx = base + (size_t)(qt * 16 + row) * DD + col;
            out[idx] = seq[idx] + mk * acc[r] * rs[row];
        }
    }
}

// ---------------------------------------------------------------------------
extern "C" void kernel_launch(void* const* d_in, const int* in_sizes, int n_in,
                              void* d_out, int out_size, void* d_ws, size_t ws_size,
                              hipStream_t stream) {
    const float* seq  = (const float*)d_in[0];
    const float* Wq   = (const float*)d_in[1];
    const float* Wk   = (const float*)d_in[2];
    const float* Wv   = (const float*)d_in[3];
    const float* mask = (const float*)d_in[4];
    float*       out  = (float*)d_out;

    // workspace: Q | K | V, each MTOT*DD floats (32 MB)
    float* Q = (float*)d_ws;
    float* K = Q + (size_t)MTOT * DD;
    float* V = K + (size_t)MTOT * DD;

    dim3 g1(DD / 16, MTOT / 16);
    qkv_proj_kernel<<<g1, 32, 0, stream>>>(seq, Wq, Wk, Wv, Q, K, V);

    const size_t shmem = (size_t)(16 * SS + 16) * sizeof(float);   // ~128 KB LDS strip
    dim3 g2(SS / 16, BB);
    attn_kernel<<<g2, 32, shmem, stream>>>(Q, K, V, seq, mask, out);
}